// MultiboxLoss_11716670783869
// MI455X (gfx1250) — compile-verified
//
#include <hip/hip_runtime.h>
#include <hip/hip_bf16.h>
#include <stdint.h>

typedef uint32_t u32;
typedef __attribute__((ext_vector_type(4))) u32 u32x4;
typedef __attribute__((ext_vector_type(8))) int i32x8;
typedef __attribute__((ext_vector_type(4))) int i32x4;

#define BB 64
#define PP 8732
#define CC 81
#define NPR 3
#define BP (BB * PP)          // 558848
#define SLAB 128              // priors per slab (one block iteration)
#define NSLABS (BP / SLAB)    // 4366 exactly
#define QPRIORS 32            // priors per wave quarter
#define QELEMS (QPRIORS * CC) // 2592 dwords per TDM

// ---------------------------------------------------------------------------
// TDM: 1-line tile (nelem dwords) from global -> LDS. D# per ISA cdna5 §8.
// amdgpu-toolchain (clang-23) builtin arity: 6 args
//   (u32x4 g0, i32x8 g1, i32x4 g2, i32x4 g3, i32x8 extra, i32 cpol)
// ---------------------------------------------------------------------------
__device__ __forceinline__ void tdm_load_1d(u32 lds_byte_off, const void* gptr, u32 nelem) {
    unsigned long long ga = (unsigned long long)(uintptr_t)gptr;
    u32 ga_lo = (u32)__builtin_amdgcn_readfirstlane((int)(u32)ga);
    u32 ga_hi = (u32)__builtin_amdgcn_readfirstlane((int)(u32)(ga >> 32));
    u32 loff  = (u32)__builtin_amdgcn_readfirstlane((int)lds_byte_off);

    u32x4 g0;
    g0[0] = 1u;                                          // count=1, user D#
    g0[1] = loff;                                        // lds_addr (bytes)
    g0[2] = ga_lo;                                       // global_addr[31:0]
    g0[3] = (ga_hi & 0x01FFFFFFu) | (2u << 30);          // global_addr[56:32] | type=2

    i32x8 g1;
    g1[0] = (int)(2u << 16);                             // wg_mask=0, data_size=2 (4B)
    g1[1] = (int)((nelem & 0xFFFFu) << 16);              // tensor_dim0[15:0]
    g1[2] = (int)((nelem >> 16) | (1u << 16));           // tensor_dim0[31:16] | tensor_dim1=1
    g1[3] = (int)((nelem & 0xFFFFu) << 16);              // tensor_dim1 hi=0 | tile_dim0=nelem
    g1[4] = 1;                                           // tile_dim1=1, tile_dim2=0
    g1[5] = (int)nelem;                                  // tensor_dim0_stride lo
    g1[6] = 0;                                           // stride hi | dim1_stride lo
    g1[7] = 0;
    i32x4 z4 = {0, 0, 0, 0};
    i32x8 z8 = {0, 0, 0, 0, 0, 0, 0, 0};
    __builtin_amdgcn_tensor_load_to_lds(g0, g1, z4, z4, z8, 0);
}

// ---------------------------------------------------------------------------
__global__ void mb_init(float* acc, u32* rowPos) {
    int t = threadIdx.x;
    if (t < 8) acc[t] = 0.0f;
    if (t < BB) rowPos[t] = 0u;
}

// ---------------------------------------------------------------------------
// Streaming pass: TDM double-buffered, barrier-free (per-wave disjoint LDS).
// ---------------------------------------------------------------------------
__global__ __launch_bounds__(SLAB) void mb_perprior(
    const float* __restrict__ conf, const float* __restrict__ pred,
    const int* __restrict__ labels, const float* __restrict__ gt,
    u32* __restrict__ keys, float* __restrict__ ceOut,
    u32* __restrict__ rowPos, float* __restrict__ acc)
{
    __shared__ float lds[2][SLAB * CC];
    const int tid  = threadIdx.x;
    const int wave = tid >> 5;
    const int lane = tid & 31;

    int slab = blockIdx.x;
    if (slab >= NSLABS) return;

    // prefetch first slab, this wave's quarter
    {
        u32 off = (u32)(uintptr_t)(&lds[0][wave * QELEMS]);
        tdm_load_1d(off, conf + (size_t)slab * SLAB * CC + (size_t)wave * QELEMS, QELEMS);
    }

    int buf = 0;
    for (; slab < NSLABS; slab += gridDim.x) {
        int nxt = slab + gridDim.x;
        if (nxt < NSLABS) {
            u32 off = (u32)(uintptr_t)(&lds[buf ^ 1][wave * QELEMS]);
            tdm_load_1d(off, conf + (size_t)nxt * SLAB * CC + (size_t)wave * QELEMS, QELEMS);
            __builtin_amdgcn_s_wait_tensorcnt(1);   // current slab's DMA done
        } else {
            __builtin_amdgcn_s_wait_tensorcnt(0);
        }

        const float* row = &lds[buf][tid * CC];     // stride 81 dwords: bank-conflict-free
        // max
        float m = row[0];
        #pragma unroll 9
        for (int c = 1; c < CC; ++c) m = fmaxf(m, row[c]);
        // sum exp
        float s = 0.0f;
        #pragma unroll 9
        for (int c = 0; c < CC; ++c) s += __expf(row[c] - m);
        float logZ = m + __logf(s);

        size_t g   = (size_t)slab * SLAB + (size_t)tid;
        int    lbl = labels[g];
        bool   pos = lbl > 0;
        float  ce  = logZ - row[lbl];
        float  bg  = logZ - row[0];

        u32 u   = __float_as_uint(bg);
        u32 key = (u & 0x80000000u) ? ~u : (u | 0x80000000u);  // descending-sortable
        if (pos) key = 0u;                                      // -inf: excluded from mining
        keys[g]  = key;
        ceOut[g] = ce;

        float posce = pos ? ce : 0.0f;
        float sl1   = 0.0f;
        if (pos) {
            atomicAdd(&rowPos[g / PP], 1u);
            float4 pv = ((const float4*)pred)[g];
            float4 gv = ((const float4*)gt)[g];
            float d;
            d = fabsf(pv.x - gv.x); sl1 += (d < 1.0f) ? 0.5f * d * d : d - 0.5f;
            d = fabsf(pv.y - gv.y); sl1 += (d < 1.0f) ? 0.5f * d * d : d - 0.5f;
            d = fabsf(pv.z - gv.z); sl1 += (d < 1.0f) ? 0.5f * d * d : d - 0.5f;
            d = fabsf(pv.w - gv.w); sl1 += (d < 1.0f) ? 0.5f * d * d : d - 0.5f;
        }
        // wave32 reduction
        #pragma unroll
        for (int o = 16; o >= 1; o >>= 1) {
            posce += __shfl_xor(posce, o, 32);
            sl1   += __shfl_xor(sl1,   o, 32);
        }
        if (lane == 0) {
            if (sl1   != 0.0f) atomicAdd(&acc[0], sl1);
            if (posce != 0.0f) atomicAdd(&acc[1], posce);
        }
        buf ^= 1;
    }
    __builtin_amdgcn_s_wait_tensorcnt(0);
}

// ---------------------------------------------------------------------------
// Per-row hard-negative selection: radix-select kth-largest key, sum CE of
// the top-k negatives. One block per batch row.
// ---------------------------------------------------------------------------
__global__ __launch_bounds__(256) void mb_select(
    const u32* __restrict__ keys, const float* __restrict__ ceIn,
    const u32* __restrict__ rowPos, float* __restrict__ acc)
{
    __shared__ u32   hist[256];
    __shared__ u32   s_prefix, s_remaining;
    __shared__ float s_sum;

    const int b   = blockIdx.x;
    const int tid = threadIdx.x;
    const u32 np  = rowPos[b];
    u32 k = np * (u32)NPR;
    if (k > (u32)PP) k = (u32)PP;
    if (k == 0) return;

    const u32*   rk = keys + (size_t)b * PP;
    const float* rc = ceIn + (size_t)b * PP;

    if (tid == 0) { s_prefix = 0u; s_remaining = k; s_sum = 0.0f; }
    __syncthreads();

    for (int shift = 24; shift >= 0; shift -= 8) {
        hist[tid] = 0u;
        __syncthreads();
        u32 prefix = s_prefix;
        u32 mask   = (shift == 24) ? 0u : (0xFFFFFFFFu << (shift + 8));
        for (int i = tid; i < PP; i += 256) {
            u32 key = rk[i];
            if ((key & mask) == (prefix & mask))
                atomicAdd(&hist[(key >> shift) & 255u], 1u);
        }
        __syncthreads();
        if (tid == 0) {
            u32 rem = s_remaining, cum = 0;
            int v = 255;
            for (; v > 0; --v) {
                if (cum + hist[v] >= rem) break;
                cum += hist[v];
            }
            s_remaining = rem - cum;
            s_prefix    = prefix | ((u32)v << shift);
        }
        __syncthreads();
    }

    const u32 kth = s_prefix;
    // strictly-greater keys are all selected negatives (positives have key 0)
    float sum = 0.0f;
    for (int i = tid; i < PP; i += 256)
        if (rk[i] > kth) sum += rc[i];
    #pragma unroll
    for (int o = 16; o >= 1; o >>= 1) sum += __shfl_xor(sum, o, 32);
    if ((tid & 31) == 0) atomicAdd(&s_sum, sum);
    __syncthreads();

    if (tid == 0) {
        // ties at the kth key: argsort is stable -> include first `remaining`
        // in index order. key==0 ties are positives (already counted).
        float t = 0.0f;
        u32 need = s_remaining;
        for (int i = 0; i < PP && need; ++i) {
            if (rk[i] == kth) { if (kth != 0u) t += rc[i]; --need; }
        }
        atomicAdd(&acc[2], s_sum + t);
    }
}

// ---------------------------------------------------------------------------
__global__ void mb_finalize(const float* acc, const u32* rowPos, float* out) {
    if (threadIdx.x == 0 && blockIdx.x == 0) {
        u32 np = 0;
        for (int b = 0; b < BB; ++b) np += rowPos[b];
        float f = (float)np;
        out[0] = acc[0] / f;              // smooth_l1_loss / num_pos
        out[1] = (acc[1] + acc[2]) / f;   // classification_loss / num_pos
    }
}

// ---------------------------------------------------------------------------
extern "C" void kernel_launch(void* const* d_in, const int* in_sizes, int n_in,
                              void* d_out, int out_size, void* d_ws, size_t ws_size,
                              hipStream_t stream) {
    const float* conf   = (const float*)d_in[0];
    const float* pred   = (const float*)d_in[1];
    const int*   labels = (const int*)d_in[2];
    const float* gt     = (const float*)d_in[3];
    float*       out    = (float*)d_out;

    char*  ws     = (char*)d_ws;
    float* acc    = (float*)ws;                              // 8 floats
    u32*   rowPos = (u32*)(ws + 64);                         // 64 u32
    u32*   keys   = (u32*)(ws + 512);                        // BP u32
    float* cew    = (float*)(ws + 512 + (size_t)BP * 4);     // BP f32

    mb_init    <<<1, 128, 0, stream>>>(acc, rowPos);
    mb_perprior<<<1024, SLAB, 0, stream>>>(conf, pred, labels, gt, keys, cew, rowPos, acc);
    mb_select  <<<BB, 256, 0, stream>>>(keys, cew, rowPos, acc);
    mb_finalize<<<1, 32, 0, stream>>>(acc, rowPos, out);
}